// my_CEloss_70471823392953
// MI455X (gfx1250) — compile-verified
//
#include <hip/hip_runtime.h>
#include <stdint.h>

// Weighted binary cross-entropy reduction, MI455X (gfx1250).
// Memory-bound: ~403 MB streamed once -> ~17us floor at 23.3 TB/s.
// Uses CDNA5 async global->LDS copies (ASYNCcnt) with double buffering,
// non-temporal hints, and wave32 shuffle reductions.

#define Bsz      128
#define Nsz      262144
#define TPB      256
#define CHUNK    8192                 // positions per block
#define STEP_POS (TPB * 4)            // 1024 positions per pipeline step
#define STEPS    (CHUNK / STEP_POS)   // 8

// Low 32 bits of a generic pointer to LDS == LDS byte offset (aperture in hi32).
__device__ __forceinline__ unsigned lds_addr32(const void* p) {
  return (unsigned)(uintptr_t)p;
}

// CDNA5 async copy: 16 bytes per lane from the global address in vaddr to the LDS address in vdst, ASYNCcnt-tracked.
__device__ __forceinline__ void async_b128(unsigned lds, const void* g) {
  asm volatile("global_load_async_to_lds_b128 %0, %1, off th:TH_LOAD_NT"
               :: "v"(lds), "v"((uint64_t)(uintptr_t)g)
               : "memory");
}

__global__ void __launch_bounds__(TPB) wbce_partial(const float* __restrict__ inp,
                                                    const int*   __restrict__ lab,
                                                    const float* __restrict__ wgt,
                                                    float*       __restrict__ ws) {
  // Per-wave-private double buffers: each thread only touches slot [buf][t].
  __shared__ float4 sp0[2][TPB];   // class-0 logits
  __shared__ float4 sp1[2][TPB];   // class-1 logits
  __shared__ int4   slb[2][TPB];   // labels

  const int t     = threadIdx.x;
  const int chunk = blockIdx.x;
  const int b     = blockIdx.y;
  const float w   = wgt[0];

  const float* p0 = inp + (size_t)b * (2 * (size_t)Nsz) + (size_t)chunk * CHUNK;
  const float* p1 = p0 + Nsz;
  const int*   lb = lab + (size_t)b * (size_t)Nsz + (size_t)chunk * CHUNK;

  // Prologue: fill buffer 0.
  async_b128(lds_addr32(&sp0[0][t]), p0 + t * 4);
  async_b128(lds_addr32(&sp1[0][t]), p1 + t * 4);
  async_b128(lds_addr32(&slb[0][t]), lb + t * 4);

  float accL = 0.0f;   // sum of per-position losses
  float accP = 0.0f;   // count of positives

  const float RLN2 = 1.44269504088896340736f;  // 1/ln2
  const float LN2  = 0.69314718055994530942f;

#pragma unroll
  for (int i = 0; i < STEPS; ++i) {
    const int cur = i & 1;
    if (i + 1 < STEPS) {
      const int nxt = cur ^ 1;
      const int off = (i + 1) * STEP_POS + t * 4;
      // Ensure our previous ds_loads from 'nxt' completed before async overwrites it.
      asm volatile("s_wait_dscnt 0" ::: "memory");
      async_b128(lds_addr32(&sp0[nxt][t]), p0 + off);
      async_b128(lds_addr32(&sp1[nxt][t]), p1 + off);
      async_b128(lds_addr32(&slb[nxt][t]), lb + off);
      // 3 newest still in flight; the 3 for 'cur' are done.
      asm volatile("s_wait_asynccnt 3" ::: "memory");
    } else {
      asm volatile("s_wait_asynccnt 0" ::: "memory");
    }

    float4 a = sp0[cur][t];
    float4 c = sp1[cur][t];
    int4   l = slb[cur][t];

    // s = softplus(d) = -log p0 ; softplus(-d) = s - d = -log p1
#define POS1(pa, pc, pl)                                                     \
    {                                                                        \
      float d = (pc) - (pa);                                                 \
      float e = __builtin_amdgcn_exp2f(-fabsf(d) * RLN2);                    \
      float s = fmaxf(d, 0.0f) + LN2 * __builtin_amdgcn_logf(1.0f + e);      \
      bool pos = ((pl) == 1);                                                \
      accL += pos ? (w * (s - d)) : s;                                       \
      accP += pos ? 1.0f : 0.0f;                                             \
    }
    POS1(a.x, c.x, l.x)
    POS1(a.y, c.y, l.y)
    POS1(a.z, c.z, l.z)
    POS1(a.w, c.w, l.w)
#undef POS1
  }

  // wave32 butterfly reduction; lane 0 of each wave commits.
  for (int off = 16; off > 0; off >>= 1) {
    accL += __shfl_xor(accL, off, 32);
    accP += __shfl_xor(accP, off, 32);
  }
  if ((t & 31) == 0) {
    atomicAdd(&ws[b], accL);
    atomicAdd(&ws[Bsz + b], accP);
  }
}

__global__ void wbce_zero(float* __restrict__ ws) {
  ws[threadIdx.x] = 0.0f;   // launched with exactly 2*Bsz threads
}

__global__ void __launch_bounds__(Bsz) wbce_final(const float* __restrict__ ws,
                                                  const float* __restrict__ wgt,
                                                  float*       __restrict__ out) {
  __shared__ float red[Bsz / 32];
  const int b   = threadIdx.x;
  const float w = wgt[0];
  float v = ws[b] / ((float)Nsz + (w - 1.0f) * ws[Bsz + b]);
  for (int off = 16; off > 0; off >>= 1) v += __shfl_xor(v, off, 32);
  if ((b & 31) == 0) red[b >> 5] = v;
  __syncthreads();
  if (b == 0) {
    float tot = 0.0f;
#pragma unroll
    for (int i = 0; i < Bsz / 32; ++i) tot += red[i];
    out[0] = tot * (1.0f / (float)Bsz);
  }
}

extern "C" void kernel_launch(void* const* d_in, const int* in_sizes, int n_in,
                              void* d_out, int out_size, void* d_ws, size_t ws_size,
                              hipStream_t stream) {
  (void)in_sizes; (void)n_in; (void)out_size; (void)ws_size;
  const float* inp = (const float*)d_in[0];   // [B, 2, N] f32
  const int*   lab = (const int*)d_in[1];     // [B, N] i32
  const float* wgt = (const float*)d_in[2];   // scalar f32
  float* ws  = (float*)d_ws;                  // [2*B] accumulators
  float* out = (float*)d_out;                 // [1]

  wbce_zero<<<1, 2 * Bsz, 0, stream>>>(ws);
  dim3 grid(Nsz / CHUNK, Bsz, 1);             // 32 x 128 blocks
  wbce_partial<<<grid, TPB, 0, stream>>>(inp, lab, wgt, ws);
  wbce_final<<<1, Bsz, 0, stream>>>(ws, wgt, out);
}